// DotPredictor_7919919693878
// MI455X (gfx1250) — compile-verified
//
#include <hip/hip_runtime.h>

// SDDMM via V_WMMA_F32_16X16X4_F32 (full-fp32 WMMA path on gfx1250).
// For each group of 16 edges: A = h[src rows] (16xK), B = h[dst rows] (Kx16),
// C += A*B over K=64 in 16 steps of 4. diag(C)[n] = <h[src[e_n]], h[dst[e_n]]>.
//
// VGPR layouts (CDNA5 ISA 7.12.2, 32-bit A 16x4 / B 4x16):
//   lane l, n = l&15, hi = l>>4, kk = k0 + 2*hi:
//     A: v0 = h[src[e_n]][kk], v1 = h[src[e_n]][kk+1]
//     B: v0 = h[dst[e_n]][kk], v1 = h[dst[e_n]][kk+1]
//   C (8 VGPRs): lanes 0-15 -> M=g,N=lane ; lanes 16-31 -> M=g+8,N=lane-16
//   diag(C)[n]: n<8 -> lane n, VGPR n ; n>=8 -> lane n+16, VGPR n-8

typedef __attribute__((ext_vector_type(2))) float v2f;
typedef __attribute__((ext_vector_type(8))) float v8f;

__global__ __launch_bounds__(256) void sddmm_wmma_f32_kernel(
    const float* __restrict__ h,
    const int*   __restrict__ src,
    const int*   __restrict__ dst,
    float*       __restrict__ out,
    int nEdges, int nGroups)
{
    const int lane          = (int)(threadIdx.x & 31u);
    const int waveInBlock   = (int)(threadIdx.x >> 5);
    const int wavesPerBlock = (int)(blockDim.x >> 5);
    const int groupStep     = (int)gridDim.x * wavesPerBlock;

    const int n     = lane & 15;   // which of the 16 edges this lane serves
    const int hi    = lane >> 4;   // half-wave select
    const int kOffs = hi * 2;      // K sub-offset per ISA layout

    for (int group = (int)blockIdx.x * wavesPerBlock + waveInBlock;
         group < nGroups; group += groupStep)
    {
        const int  e     = group * 16 + n;
        const bool valid = (e < nEdges);
        const int  ec    = valid ? e : 0;

        const long long srow = (long long)src[ec];
        const long long drow = (long long)dst[ec];
        const float* __restrict__ aRow = h + srow * 64;
        const float* __restrict__ bRow = h + drow * 64;

        v8f c = {};
        // Uniform K loop: EXEC stays all-1s across every WMMA (ISA requirement).
#pragma unroll
        for (int k0 = 0; k0 < 64; k0 += 4) {
            const int kk = k0 + kOffs;
            v2f a = *(const v2f*)(aRow + kk);   // 8B-aligned gather (L2-resident h)
            v2f b = *(const v2f*)(bRow + kk);
            c = __builtin_amdgcn_wmma_f32_16x16x4_f32(
                    /*neg_a=*/false, a, /*neg_b=*/false, b,
                    /*c_mod=*/(short)0, c, /*reuse_a=*/false, /*reuse_b=*/false);
        }

        // Extract diag(C) with branchless unrolled selects.
        float diag = 0.0f;
#pragma unroll
        for (int g = 0; g < 8; ++g) {
            const bool sel = hi ? ((lane - 24) == g) : (lane == g);
            diag = sel ? (float)c[g] : diag;
        }

        const bool doStore = valid && ((hi == 0) ? (lane < 8) : (lane >= 24));
        if (doStore) {
            out[group * 16 + (hi ? (lane - 16) : lane)] = diag;
        }
    }
}

extern "C" void kernel_launch(void* const* d_in, const int* in_sizes, int n_in,
                              void* d_out, int out_size, void* d_ws, size_t ws_size,
                              hipStream_t stream) {
    const float* h   = (const float*)d_in[0];
    const int*   src = (const int*)d_in[1];
    const int*   dst = (const int*)d_in[2];
    float*       out = (float*)d_out;

    const int nEdges  = in_sizes[1];
    const int nGroups = (nEdges + 15) / 16;   // 78,125 for E=1.25M

    const int block         = 256;            // 8 wave32s per block
    const int wavesPerBlock = block / 32;
    int grid = (nGroups + wavesPerBlock - 1) / wavesPerBlock;  // ~9766
    if (grid < 1) grid = 1;

    sddmm_wmma_f32_kernel<<<grid, block, 0, stream>>>(h, src, dst, out,
                                                      nEdges, nGroups);
}